// spikegelu0_30_20684562497587
// MI455X (gfx1250) — compile-verified
//
#include <hip/hip_runtime.h>

// Elementwise 16-step SNN scan: out = f(x) per element.
// Division eliminated analytically: sign((v-T)/(|v|+1)) == sign(v-T) since |v|+1 > 0.
// Pure streaming kernel -> 128-bit nontemporal loads/stores, unrolled register scan.
// FMAs expressed on f4 vectors so the backend can emit V_PK_FMA_F32 (2xf32/instr).

typedef float f4 __attribute__((ext_vector_type(4)));

__device__ __constant__ float cH[16] = {
    -0.73437643f, 3.319645f,   2.1290164f,  3.6901689f,
    3.302721f,    3.2154958f,  2.9151256f,  3.1718695f,
    3.1084518f,   2.6179547f,  2.235003f,   1.2864777f,
    0.52327204f, -0.08344932f, -1.8249638f, 3.0859442f};

__device__ __constant__ float cD[16] = {
    3.3848417f,  -0.07072583f, 3.691287f,   3.3074934f,
    3.2120926f,   2.9196491f,  3.1727686f,  3.1054153f,
    2.621943f,    2.2355895f,  1.2878408f,  0.52468026f,
    0.12632068f,  0.14482783f, 0.3153498f,  0.13054803f};

__device__ __constant__ float cT[16] = {
    3.1538513f,  -1.0211663f,  1.3714716f,  1.0718397f,
    1.0049332f,   0.68078244f, 1.0151638f,  0.8858697f,
    0.4037103f,   0.01490025f, -0.90781677f, -1.6859558f,
    -1.9241625f, -2.0441303f,  0.279356f,   0.1315174f};

__device__ __forceinline__ f4 splat4(float s) { return (f4){s, s, s, s}; }

// Vectorized scan over 4 independent elements.
// Step 0 peeled: z starts at 0, so v is unchanged and out = (x>T0)?d0:0.
// z in {0,1} makes every fma/mul bit-exact vs the reference's sub/add forms.
__device__ __forceinline__ f4 snn_scan_vec4(f4 x) {
  f4 v = x;
  f4 z;
#pragma unroll
  for (int j = 0; j < 4; ++j) z[j] = (v[j] > cT[0]) ? 1.0f : 0.0f;
  f4 o = z * splat4(cD[0]);
#pragma unroll
  for (int t = 1; t < 16; ++t) {
    v = __builtin_elementwise_fma(z, splat4(-cH[t]), v);   // v = v - z*h_t (pk_fma)
#pragma unroll
    for (int j = 0; j < 4; ++j)
      z[j] = (v[j] > cT[t]) ? 1.0f : 0.0f;                 // spike: sign(v - T_t)
    o = __builtin_elementwise_fma(z, splat4(cD[t]), o);    // out += z*d_t (pk_fma)
  }
  return o;
}

__device__ __forceinline__ float snn_scan_elem(float x0) {
  float v = x0;
  float z = (v > cT[0]) ? 1.0f : 0.0f;
  float o = z * cD[0];
#pragma unroll
  for (int t = 1; t < 16; ++t) {
    v = __builtin_fmaf(z, -cH[t], v);
    z = (v > cT[t]) ? 1.0f : 0.0f;
    o = __builtin_fmaf(z, cD[t], o);
  }
  return o;
}

#define BLOCK 256
#define V4_PER_THREAD 4

// Fast path: grid exactly covers the data, no bounds checks, single base
// address + immediate offsets -> clause of 4 b128 NT loads, 4 b128 NT stores.
__global__ __launch_bounds__(BLOCK) void snn_scan_v4_exact(const f4* __restrict__ xv,
                                                           f4* __restrict__ ov) {
  const long long base =
      (long long)blockIdx.x * (BLOCK * V4_PER_THREAD) + threadIdx.x;
  const f4* __restrict__ xp = xv + base;
  f4* __restrict__ op = ov + base;

  f4 in[V4_PER_THREAD];
#pragma unroll
  for (int k = 0; k < V4_PER_THREAD; ++k)
    in[k] = __builtin_nontemporal_load(xp + k * BLOCK);

#pragma unroll
  for (int k = 0; k < V4_PER_THREAD; ++k) {
    f4 r = snn_scan_vec4(in[k]);
    __builtin_nontemporal_store(r, op + k * BLOCK);
  }
}

// Generic guarded path (only used if the size doesn't divide evenly).
__global__ __launch_bounds__(BLOCK) void snn_scan_v4_guarded(const f4* __restrict__ xv,
                                                             f4* __restrict__ ov,
                                                             long long n4) {
  const long long base =
      (long long)blockIdx.x * (BLOCK * V4_PER_THREAD) + threadIdx.x;
#pragma unroll
  for (int k = 0; k < V4_PER_THREAD; ++k) {
    long long i = base + (long long)k * BLOCK;
    if (i < n4) {
      f4 in = __builtin_nontemporal_load(xv + i);
      f4 r = snn_scan_vec4(in);
      __builtin_nontemporal_store(r, ov + i);
    }
  }
}

__global__ __launch_bounds__(BLOCK) void snn_scan_tail(const float* __restrict__ x,
                                                       float* __restrict__ out,
                                                       long long start,
                                                       long long n) {
  long long i = start + (long long)blockIdx.x * BLOCK + threadIdx.x;
  if (i < n) out[i] = snn_scan_elem(x[i]);
}

extern "C" void kernel_launch(void* const* d_in, const int* in_sizes, int n_in,
                              void* d_out, int out_size, void* d_ws, size_t ws_size,
                              hipStream_t stream) {
  const float* x = (const float*)d_in[0];
  float* out = (float*)d_out;

  const long long n = (long long)in_sizes[0];   // 16*4096*1024 = 67,108,864
  const long long n4 = n >> 2;                  // float4 groups
  const long long per_block = (long long)BLOCK * V4_PER_THREAD;

  if (n4 > 0) {
    if ((n4 % per_block) == 0) {
      const long long blocks = n4 / per_block;  // 16384 for this shape
      snn_scan_v4_exact<<<(unsigned)blocks, BLOCK, 0, stream>>>(
          (const f4*)x, (f4*)out);
    } else {
      const long long blocks = (n4 + per_block - 1) / per_block;
      snn_scan_v4_guarded<<<(unsigned)blocks, BLOCK, 0, stream>>>(
          (const f4*)x, (f4*)out, n4);
    }
  }

  const long long rem = n - (n4 << 2);          // 0 for this shape
  if (rem > 0) {
    const long long rblocks = (rem + BLOCK - 1) / BLOCK;
    snn_scan_tail<<<(unsigned)rblocks, BLOCK, 0, stream>>>(x, out, n4 << 2, n);
  }
}